// NodeConv_53644141527058
// MI455X (gfx1250) — compile-verified
//
#include <hip/hip_runtime.h>
#include <hip/hip_bf16.h>

#define N_NODES   100000
#define N_EDGES   1600000
#define IN_CH     128
#define OUT_CH    128
#define EDGE_DIM  6
#define STATE_DIM 5

typedef __attribute__((ext_vector_type(2))) float v2f;
typedef __attribute__((ext_vector_type(8))) float v8f;

__device__ __forceinline__ float elu1(float v) {
    // jax.nn.elu with alpha=1
    return v > 0.0f ? v : (__expf(v) - 1.0f);
}

// ---- gfx1250 async global->LDS copies (ASYNCcnt-tracked) --------------------
// LDS destination address = low 32 bits of the generic pointer (aperture||offset).
__device__ __forceinline__ void async_g2l_b32(const float* gptr, float* ldsptr) {
    unsigned lds = (unsigned)(unsigned long long)ldsptr;
    asm volatile("global_load_async_to_lds_b32 %0, %1, off"
                 :: "v"(lds), "v"(gptr) : "memory");
}
__device__ __forceinline__ void async_g2l_b128(const float* gptr, float* ldsptr) {
    unsigned lds = (unsigned)(unsigned long long)ldsptr;
    asm volatile("global_load_async_to_lds_b128 %0, %1, off"
                 :: "v"(lds), "v"(gptr) : "memory");
}
__device__ __forceinline__ void wait_async0() {
#if __has_builtin(__builtin_amdgcn_s_wait_asynccnt)
    __builtin_amdgcn_s_wait_asynccnt(0);
#else
    asm volatile("s_wait_asynccnt 0x0" ::: "memory");
#endif
}

// LDS geometry shared by both kernels.
//  - W is stored TRANSPOSED: sWt[n * KSTR + k]  (k contiguous -> B frags use
//    immediate offsets from a single base, like the A row).
//  - KSTR = 149: 21 mod 64, gcd 1 -> B-fragment reads are bank-conflict free.
//  - ASTR = 148: rows 16B-aligned for B128 async writes; 4*(5m mod 16) bank
//    mapping -> A-fragment reads conflict-free.
#define KSTR 149
#define ASTR 148

// -----------------------------------------------------------------------------
// Kernel 1: d_out = elu(concat(x, gstate) @ W_root + b_root)   [N, 128]
// K = 133 padded to 136 (34 WMMA k-steps). Async double-buffered A tiles.
// -----------------------------------------------------------------------------
__global__ void __launch_bounds__(256, 1)
node_root_kernel(const float* __restrict__ x,
                 const float* __restrict__ gstate,
                 const float* __restrict__ Wroot,   // [133,128] row-major
                 const float* __restrict__ broot,   // [128]
                 float* __restrict__ out)           // [N,128]
{
    constexpr int KREAL = IN_CH + STATE_DIM;   // 133
    constexpr int KP    = 136;                 // padded K (WMMA loop bound)

    extern __shared__ float smem[];
    float* sWt   = smem;                // 128 * KSTR (transposed W)
    float* sA0   = sWt + 128 * KSTR;    // 16 * ASTR
    float* sA1   = sA0 + 16 * ASTR;     // 16 * ASTR
    float* sBias = sA1 + 16 * ASTR;     // 128

    const int tid  = threadIdx.x;
    const int wave = tid >> 5;
    const int lane = tid & 31;
    const int m    = lane & 15;
    const int half = lane >> 4;

    // Stage W_root transposed (zero-pad k in [133,136)); zero both A buffers.
    for (int i = tid; i < KP * OUT_CH; i += 256) {
        int k = i >> 7, n = i & 127;                  // coalesced in n
        sWt[n * KSTR + k] = (k < KREAL) ? Wroot[k * OUT_CH + n] : 0.0f;
    }
    for (int i = tid; i < 2 * 16 * ASTR; i += 256) sA0[i] = 0.0f;
    if (tid < OUT_CH) sBias[tid] = broot[tid];
    __syncthreads();

    const int ntiles = N_NODES / 16;   // 6250, exact

    auto issue_gather = [&](int t, float* buf) {
        const int nd0 = t * 16;
        for (int i = tid; i < 16 * 32; i += 256) {    // x rows: B128 chunks
            int r = i >> 5, c = (i & 31) << 2;
            async_g2l_b128(x + (size_t)(nd0 + r) * IN_CH + c, buf + r * ASTR + c);
        }
        for (int i = tid; i < 16 * STATE_DIM; i += 256) {  // gstate tail: B32
            int r = i / STATE_DIM, f2 = i - r * STATE_DIM;
            async_g2l_b32(gstate + (size_t)(nd0 + r) * STATE_DIM + f2,
                          buf + r * ASTR + IN_CH + f2);
        }
    };

    int t = blockIdx.x;                 // gridDim.x < ntiles, so first issue is valid
    issue_gather(t, sA0);
    int bufsel = 0;

    for (; t < ntiles; t += gridDim.x) {
        float* cur = bufsel ? sA1 : sA0;
        float* nxt = bufsel ? sA0 : sA1;

        wait_async0();
        __syncthreads();                // all waves done reading 'nxt' & loads landed

        int tn = t + gridDim.x;
        if (tn < ntiles) issue_gather(tn, nxt);

        // Each wave: 16x16 tile, rows [t*16..+16), cols [wave*16..+16).
        const int n0 = wave * 16;
        const float* aRow = cur + m * ASTR + 2 * half;
        const float* bRow = sWt + (n0 + m) * KSTR + 2 * half;

        v8f acc = {};
        #pragma unroll
        for (int k = 0; k < KP; k += 4) {
            v2f a; a.x = aRow[k]; a.y = aRow[k + 1];
            v2f b; b.x = bRow[k]; b.y = bRow[k + 1];
            acc = __builtin_amdgcn_wmma_f32_16x16x4_f32(
                false, a, false, b, (short)0, acc, false, false);
        }

        const int col  = n0 + m;
        const float bb = sBias[col];
        #pragma unroll
        for (int i = 0; i < 8; ++i) {
            int r = half * 8 + i;           // C VGPR i -> row 8*half+i
            out[(size_t)(t * 16 + r) * OUT_CH + col] = elu1(acc[i] + bb);
        }
        bufsel ^= 1;
    }
}

// -----------------------------------------------------------------------------
// Kernel 2: per 16-edge tile: elu(concat(xs[src], edge_attr) @ W_neg + b_neg),
//           atomicAdd into out[dst]. K = 139 padded to 144 (36 k-steps).
// -----------------------------------------------------------------------------
__global__ void __launch_bounds__(256, 1)
edge_mlp_scatter_kernel(const float* __restrict__ x,
                        const int*   __restrict__ eidx,    // [2, E] (row0=dst, row1=src)
                        const float* __restrict__ eattr,   // [E, 6]
                        const float* __restrict__ gstate,  // [N, 5]
                        const float* __restrict__ Wneg,    // [139,128] row-major
                        const float* __restrict__ bneg,    // [128]
                        float* __restrict__ out)           // [N,128], accumulated
{
    constexpr int KREAL = IN_CH + STATE_DIM + EDGE_DIM;    // 139
    constexpr int KP    = 144;
    constexpr int TAIL  = STATE_DIM + EDGE_DIM;            // 11

    extern __shared__ float smem[];
    float* sWt   = smem;                // 128 * KSTR (transposed W)
    float* sA0   = sWt + 128 * KSTR;    // 16 * ASTR
    float* sA1   = sA0 + 16 * ASTR;     // 16 * ASTR
    float* sBias = sA1 + 16 * ASTR;     // 128

    const int tid  = threadIdx.x;
    const int wave = tid >> 5;
    const int lane = tid & 31;
    const int m    = lane & 15;
    const int half = lane >> 4;

    for (int i = tid; i < KP * OUT_CH; i += 256) {
        int k = i >> 7, n = i & 127;
        sWt[n * KSTR + k] = (k < KREAL) ? Wneg[k * OUT_CH + n] : 0.0f;
    }
    for (int i = tid; i < 2 * 16 * ASTR; i += 256) sA0[i] = 0.0f;
    if (tid < OUT_CH) sBias[tid] = bneg[tid];
    __syncthreads();

    const int ntiles = N_EDGES / 16;    // 100000, exact

    auto issue_gather = [&](int t, float* buf) {
        const int e0 = t * 16;
        // xs[src] x-part: random row gather (L2-resident), B128 chunks
        for (int i = tid; i < 16 * 32; i += 256) {
            int r = i >> 5, c = (i & 31) << 2;
            int s = eidx[N_EDGES + e0 + r];           // src node
            async_g2l_b128(x + (size_t)s * IN_CH + c, buf + r * ASTR + c);
        }
        // tail: 16 rows x (5 gstate + 6 edge_attr), B32
        for (int i = tid; i < 16 * TAIL; i += 256) {
            int r = i / TAIL, f2 = i - r * TAIL;
            int e = e0 + r;
            const float* g;
            if (f2 < STATE_DIM) {
                int s = eidx[N_EDGES + e];
                g = gstate + (size_t)s * STATE_DIM + f2;
            } else {
                g = eattr + (size_t)e * EDGE_DIM + (f2 - STATE_DIM);
            }
            async_g2l_b32(g, buf + r * ASTR + IN_CH + f2);
        }
    };

    int t = blockIdx.x;
    issue_gather(t, sA0);
    int bufsel = 0;

    for (; t < ntiles; t += gridDim.x) {
        float* cur = bufsel ? sA1 : sA0;
        float* nxt = bufsel ? sA0 : sA1;

        wait_async0();
        __syncthreads();

        int tn = t + gridDim.x;
        if (tn < ntiles) issue_gather(tn, nxt);

        const int n0 = wave * 16;
        const float* aRow = cur + m * ASTR + 2 * half;
        const float* bRow = sWt + (n0 + m) * KSTR + 2 * half;

        v8f acc = {};
        #pragma unroll
        for (int k = 0; k < KP; k += 4) {
            v2f a; a.x = aRow[k]; a.y = aRow[k + 1];
            v2f b; b.x = bRow[k]; b.y = bRow[k + 1];
            acc = __builtin_amdgcn_wmma_f32_16x16x4_f32(
                false, a, false, b, (short)0, acc, false, false);
        }

        const int e0   = t * 16;
        const int col  = n0 + m;
        const float bb = sBias[col];
        #pragma unroll
        for (int i = 0; i < 8; ++i) {
            int r = half * 8 + i;
            int d = eidx[e0 + r];                     // dst node (segment id)
            atomicAdd(&out[(size_t)d * OUT_CH + col], elu1(acc[i] + bb));
        }
        bufsel ^= 1;
    }
}

// -----------------------------------------------------------------------------
extern "C" void kernel_launch(void* const* d_in, const int* in_sizes, int n_in,
                              void* d_out, int out_size, void* d_ws, size_t ws_size,
                              hipStream_t stream) {
    const float* x     = (const float*)d_in[0];
    const int*   eidx  = (const int*)  d_in[1];
    const float* eattr = (const float*)d_in[2];
    const float* gs    = (const float*)d_in[3];
    const float* Wneg  = (const float*)d_in[4];
    const float* bneg  = (const float*)d_in[5];
    const float* Wroot = (const float*)d_in[6];
    const float* broot = (const float*)d_in[7];
    float* out = (float*)d_out;

    const size_t shm = (size_t)(128 * KSTR + 2 * 16 * ASTR + 128) * sizeof(float);

    // Allow >64KB dynamic LDS (CDNA5 WGP has 320KB).
    (void)hipFuncSetAttribute((const void*)node_root_kernel,
                              hipFuncAttributeMaxDynamicSharedMemorySize, (int)shm);
    (void)hipFuncSetAttribute((const void*)edge_mlp_scatter_kernel,
                              hipFuncAttributeMaxDynamicSharedMemorySize, (int)shm);

    // 1) Initialize out with the root/self term (covers every output element).
    node_root_kernel<<<512, 256, shm, stream>>>(x, gs, Wroot, broot, out);

    // 2) Edge MLP + scatter-add (stream-ordered after kernel 1).
    edge_mlp_scatter_kernel<<<1536, 256, shm, stream>>>(
        x, eidx, eattr, gs, Wneg, bneg, out);
}